// LIIF_ONNX_82884278878344
// MI455X (gfx1250) — compile-verified
//
#include <hip/hip_runtime.h>
#include <hip/hip_bf16.h>

// LIIF: feat (1,64,128,128) f32, coord (1,Q,2), cell (1,Q,2), W (580,3), b(3)

typedef __attribute__((ext_vector_type(2))) float v2f_;
typedef __attribute__((ext_vector_type(8))) float v8f_;

#define HH 128
#define WW 128
#define CC 64
#define KTOT 576            // C*9
#define NPIX (HH*WW)        // 16384
#define NQ   (512*512)      // 262144

#define CPH 32              // channels per LDS phase
#define ROWP 130            // padded LDS row: [0]=left zero, [1..128]=x, [129]=right zero
#define LF_FLOATS (CPH*3*ROWP)   // 12480
#define LW_FLOATS 1792           // 576*3 = 1728, padded
// total LDS = (12480+1792)*4 = 57,088 B  (< 64 KB static; 2 blocks/WGP possible)

// ---------------------------------------------------------------------------
// Stage 1: T[pix][o] = unfold3(feat)[:,pix] @ W[:576,o]  via fp32 WMMA,
// LDS-staged operands, 2 channel-phases of 32, prefetch of phase-1 data
// overlapped with phase-0 compute.
//
// Fragment layouts (CDNA5 ISA 7.12.2, 32-bit, wave32):
//   A 16x4:  lane m=lane&15; half=lane>>4; a[v] = A[m, half*2+v]
//   B 4x16:  lane n=lane&15; half=lane>>4; b[v] = B[half*2+v, n]
//   C/D 16x16: vgpr j; lanes0-15: M=j, N=lane; lanes16-31: M=j+8, N=lane-16
// ---------------------------------------------------------------------------
__global__ __launch_bounds__(256)
void liif_conv_wmma(const float* __restrict__ feat,   // (64,128,128)
                    const float* __restrict__ Wm,     // (580,3)
                    float* __restrict__ T)            // (16384,4) padded
{
    __shared__ float lf[LF_FLOATS];
    __shared__ float lw[LW_FLOATS];

    const int tid  = threadIdx.x;
    const int wave = tid >> 5;        // 0..7
    const int lane = tid & 31;
    const int half = lane >> 4;       // 0 or 1
    const int m    = lane & 15;       // A: pixel-in-tile; B/D: output column

    const int y  = blockIdx.x;        // image row, 0..127
    const int x0 = wave * 16;         // tile start within row
    const int p0 = y * WW + x0;       // first pixel of this wave's tile

    v8f_ acc = {0.f, 0.f, 0.f, 0.f, 0.f, 0.f, 0.f, 0.f};

    for (int ph = 0; ph < 2; ++ph) {
        const int cbase = ph * CPH;

        // ---- stage feat rows y-1..y+1, channels [cbase, cbase+32) into LDS
        for (int idx = tid; idx < CPH * 3 * WW; idx += 256) {
            const int cl  = idx / (3 * WW);
            const int rem = idx - cl * (3 * WW);
            const int r   = rem >> 7;           // 0..2
            const int x   = rem & 127;
            const int yy  = y + r - 1;
            float v = 0.f;
            if ((unsigned)yy < (unsigned)HH)
                v = feat[(cbase + cl) * NPIX + (yy << 7) + x];
            lf[cl * (3 * ROWP) + r * ROWP + x + 1] = v;
        }
        // zero the left/right border cells
        for (int idx = tid; idx < CPH * 3 * 2; idx += 256) {
            const int pair = idx >> 1;                 // cl*3 + r
            const int pos  = (idx & 1) ? (ROWP - 1) : 0;
            lf[pair * ROWP + pos] = 0.f;
        }
        // ---- stage W once
        if (ph == 0) {
            for (int idx = tid; idx < LW_FLOATS; idx += 256)
                lw[idx] = (idx < KTOT * 3) ? Wm[idx] : 0.f;
        }
        __syncthreads();

        // ---- while phase-0 computes, prefetch phase-1's feat lines
        //      (32 ch x 3 rows x 4 cache lines of 128 B) -> global_prefetch_b8
        if (ph == 0) {
            for (int i = tid; i < CPH * 3 * 4; i += 256) {
                const int cl  = i / 12;
                const int rem = i - cl * 12;
                const int r   = rem >> 2;
                const int seg = rem & 3;
                const int yy  = y + r - 1;
                if ((unsigned)yy < (unsigned)HH)
                    __builtin_prefetch(
                        &feat[(CPH + cl) * NPIX + (yy << 7) + seg * 32], 0, 1);
            }
        }

        // ---- K loop over this phase's 288 k-values (LDS-fed WMMA)
#pragma unroll 4
        for (int k0 = ph * 288; k0 < ph * 288 + 288; k0 += 4) {
            v2f_ a, b;
#pragma unroll
            for (int v = 0; v < 2; ++v) {
                const int k   = k0 + half * 2 + v;    // global k, 0..575
                const int c   = k / 9;
                const int rem = k - c * 9;
                const int di  = rem / 3;
                const int dj  = rem - di * 3;
                const int cl  = c - cbase;
                // A[m,k] = padded feat: LDS index (x0+m+dj-1)+1
                a[v] = lf[cl * (3 * ROWP) + di * ROWP + (x0 + m + dj)];
                // B[k,n]: valid only for n<3
                const float t = lw[k * 3 + m];        // k*3+m <= 1740 < 1792
                b[v] = (m < 3) ? t : 0.f;
            }
            acc = __builtin_amdgcn_wmma_f32_16x16x4_f32(
                false, a, false, b, (short)0, acc, false, false);
        }
        __syncthreads();   // safe to re-stage LDS for next phase
    }

    // ---- write the 3 valid output columns
    if (m < 3) {
#pragma unroll
        for (int j = 0; j < 8; ++j) {
            const int pix = p0 + j + half * 8;
            T[pix * 4 + m] = acc[j];
        }
    }
}

// ---------------------------------------------------------------------------
// Stage 2: per-query table lookup + tail dot + area blend (memory-bound).
// ---------------------------------------------------------------------------
__global__ __launch_bounds__(256)
void liif_query(const float* __restrict__ coord,   // (Q,2)
                const float* __restrict__ cell,    // (Q,2)
                const float* __restrict__ Wm,      // (580,3)
                const float* __restrict__ bias,    // (3,)
                const float* __restrict__ T,       // (16384,4)
                float* __restrict__ out)           // (Q,3)
{
    const int q = blockIdx.x * blockDim.x + threadIdx.x;
    if (q >= NQ) return;

    const float2 co = ((const float2*)coord)[q];
    const float2 ce = ((const float2*)cell)[q];
    const float cx = co.x, cy = co.y;
    const float rcx = ce.x * (float)HH;
    const float rcy = ce.y * (float)WW;

    float w576[3], w577[3], base[3];
#pragma unroll
    for (int o = 0; o < 3; ++o) {
        w576[o] = Wm[576 * 3 + o];
        w577[o] = Wm[577 * 3 + o];
        base[o] = rcx * Wm[578 * 3 + o] + rcy * Wm[579 * 3 + o] + bias[o];
    }

    float areas[4];
    float partial[4][3];

    int s = 0;
#pragma unroll
    for (int vx = -1; vx <= 1; vx += 2) {
#pragma unroll
        for (int vy = -1; vy <= 1; vy += 2) {
            const float sx = (float)vx * (1.0f / HH) + 1e-6f;
            const float sy = (float)vy * (1.0f / WW) + 1e-6f;
            float gx = fminf(fmaxf(cx + sx, -1.0f + 1e-6f), 1.0f - 1e-6f);
            float gy = fminf(fmaxf(cy + sy, -1.0f + 1e-6f), 1.0f - 1e-6f);
            int ih = (int)floorf((gx + 1.0f) * (HH * 0.5f));
            int iw = (int)floorf((gy + 1.0f) * (WW * 0.5f));
            ih = min(max(ih, 0), HH - 1);
            iw = min(max(iw, 0), WW - 1);
            const float qx = -1.0f + (2.0f * (float)ih + 1.0f) * (1.0f / HH);
            const float qy = -1.0f + (2.0f * (float)iw + 1.0f) * (1.0f / WW);
            const float rx = (cx - qx) * (float)HH;
            const float ry = (cy - qy) * (float)WW;
            areas[s] = fabsf(rx * ry) + 1e-9f;

            const float4 t = ((const float4*)T)[ih * WW + iw];
            partial[s][0] = t.x + rx * w576[0] + ry * w577[0];
            partial[s][1] = t.y + rx * w576[1] + ry * w577[1];
            partial[s][2] = t.z + rx * w576[2] + ry * w577[2];
            ++s;
        }
    }

    const float tot = areas[0] + areas[1] + areas[2] + areas[3];
    const float inv = 1.0f / tot;
    const float wgt[4] = {areas[3] * inv, areas[2] * inv,
                          areas[1] * inv, areas[0] * inv};

#pragma unroll
    for (int o = 0; o < 3; ++o) {
        float r = base[o];  // constant tail factors out: blend weights sum to 1
#pragma unroll
        for (int t2 = 0; t2 < 4; ++t2) r += wgt[t2] * partial[t2][o];
        out[q * 3 + o] = r;
    }
}

extern "C" void kernel_launch(void* const* d_in, const int* in_sizes, int n_in,
                              void* d_out, int out_size, void* d_ws, size_t ws_size,
                              hipStream_t stream) {
    const float* feat  = (const float*)d_in[0];
    const float* coord = (const float*)d_in[1];
    const float* cell  = (const float*)d_in[2];
    const float* Wm    = (const float*)d_in[3];
    const float* bias  = (const float*)d_in[4];
    float* out = (float*)d_out;
    float* T   = (float*)d_ws;   // needs 16384*4*4 = 256 KB scratch

    // Stage 1: one block per image row; 8 waves/block, one 16-pixel tile each.
    liif_conv_wmma<<<HH, 256, 0, stream>>>(feat, Wm, T);
    // Stage 2: one thread per query.
    liif_query<<<NQ / 256, 256, 0, stream>>>(coord, cell, Wm, bias, T, out);
}